// MoERouter_40407052320886
// MI455X (gfx1250) — compile-verified
//
#include <hip/hip_runtime.h>
#include <hip/hip_bf16.h>
#include <math.h>

typedef float v2f __attribute__((ext_vector_type(2)));
typedef float v4f __attribute__((ext_vector_type(4)));
typedef float v8f __attribute__((ext_vector_type(8)));

#define NROWS      8192
#define KDIM       7168
#define NEXP       256
#define NGROUP     8
#define EPG        32
#define TOPK       8
#define TOPKG      4

#define TILE_ROWS  64      // rows per block
#define KB         32      // K panel width staged in LDS
#define NPANELS    (KDIM / KB)
#define A_STRIDE   36      // padded LDS row stride for A (floats), 16B aligned, conflict-free
#define B_STRIDE   272     // padded LDS row stride for B (floats), +32-bank shift for hi half

#if __has_builtin(__builtin_amdgcn_global_load_async_to_lds_b128)
#define HAVE_ASYNC_LDS 1
#else
#define HAVE_ASYNC_LDS 0
#endif

#if HAVE_ASYNC_LDS
// Parameter types per hipcc diagnostic: arg0 is AS1 pointer to
// 'int __attribute__((vector_size(16)))' (printed as "__device__ *").
typedef int gv4i __attribute__((vector_size(16)));
typedef __attribute__((address_space(1))) gv4i* global_v4i_ptr;
typedef __attribute__((address_space(3))) gv4i* lds_v4i_ptr;

__device__ __forceinline__ void async_copy16(const float* g, float* l) {
    // GLOBAL_LOAD_ASYNC_TO_LDS_B128: per-lane 16B global -> LDS, tracked by ASYNCcnt
    __builtin_amdgcn_global_load_async_to_lds_b128(
        (global_v4i_ptr)(g),
        (lds_v4i_ptr)(l),
        /*imm offset*/0, /*cpol*/0);
}
__device__ __forceinline__ void wait_async0() {
#if __has_builtin(__builtin_amdgcn_s_wait_asynccnt)
    __builtin_amdgcn_s_wait_asynccnt(0);
#else
    asm volatile("s_wait_asynccnt 0x0" ::: "memory");
#endif
}
#endif

// ---------------------------------------------------------------------------
// Kernel 1: scores = sigmoid(x @ W), fp32 via V_WMMA_F32_16X16X4_F32
// Double-buffered LDS panels; async global->LDS when available.
// ---------------------------------------------------------------------------
__global__ __launch_bounds__(256) void moe_gemm_sigmoid_kernel(
    const float* __restrict__ x, const float* __restrict__ W,
    float* __restrict__ scores)
{
    __shared__ float As[2][TILE_ROWS * A_STRIDE];  // 2 x 64x36 floats  (18,432 B)
    __shared__ float Bs[2][KB * B_STRIDE];         // 2 x 32x272 floats (69,632 B)

    const int row0  = blockIdx.x * TILE_ROWS;
    const int tid   = threadIdx.x;
    const int wave  = tid >> 5;         // 0..7
    const int lane  = tid & 31;
    const int lhalf = lane >> 4;        // 0/1
    const int lmod  = lane & 15;
    const int col0  = wave * 32;        // each wave covers 2 col tiles (32 cols)

    v8f acc[4][2];
    #pragma unroll
    for (int rt = 0; rt < 4; ++rt)
        #pragma unroll
        for (int ct = 0; ct < 2; ++ct)
            acc[rt][ct] = (v8f){0.f,0.f,0.f,0.f,0.f,0.f,0.f,0.f};

#if HAVE_ASYNC_LDS
    // ---- async double-buffered pipeline: one barrier per panel ----
    {   // prologue: issue panel 0 into buffer 0
        #pragma unroll
        for (int i = tid; i < (TILE_ROWS * KB / 4); i += 256) {
            const int r = i >> 3, c4 = (i & 7) * 4;
            async_copy16(x + (size_t)(row0 + r) * KDIM + c4, &As[0][r * A_STRIDE + c4]);
        }
        #pragma unroll
        for (int i = tid; i < (KB * NEXP / 4); i += 256) {
            const int r = i >> 6, c4 = (i & 63) * 4;
            async_copy16(W + (size_t)r * NEXP + c4, &Bs[0][r * B_STRIDE + c4]);
        }
    }
    for (int p = 0; p < NPANELS; ++p) {
        const int buf = p & 1;
        wait_async0();       // this wave's panel-p asyncs are in LDS
        __syncthreads();     // all waves: panel-p staged AND panel-(p-1) compute done
        if (p + 1 < NPANELS) {
            const int k0n = (p + 1) * KB;
            #pragma unroll
            for (int i = tid; i < (TILE_ROWS * KB / 4); i += 256) {
                const int r = i >> 3, c4 = (i & 7) * 4;
                async_copy16(x + (size_t)(row0 + r) * KDIM + k0n + c4,
                             &As[buf ^ 1][r * A_STRIDE + c4]);
            }
            #pragma unroll
            for (int i = tid; i < (KB * NEXP / 4); i += 256) {
                const int r = i >> 6, c4 = (i & 63) * 4;
                async_copy16(W + (size_t)(k0n + r) * NEXP + c4,
                             &Bs[buf ^ 1][r * B_STRIDE + c4]);
            }
        }
        const float* Ab = As[buf];
        const float* Bb = Bs[buf];
#else
    // ---- fallback: register staging, two barriers per panel ----
    for (int p = 0; p < NPANELS; ++p) {
        const int buf = p & 1;
        const int k0 = p * KB;
        __syncthreads();
        #pragma unroll
        for (int i = tid; i < (TILE_ROWS * KB / 4); i += 256) {
            const int r = i >> 3, c4 = (i & 7) * 4;
            v4f v = __builtin_nontemporal_load(
                (const v4f*)(x + (size_t)(row0 + r) * KDIM + k0 + c4));
            float* pp = &As[buf][r * A_STRIDE + c4];
            pp[0] = v[0]; pp[1] = v[1]; pp[2] = v[2]; pp[3] = v[3];
        }
        #pragma unroll
        for (int i = tid; i < (KB * NEXP / 4); i += 256) {
            const int r = i >> 6, c4 = (i & 63) * 4;
            v4f v = *(const v4f*)(W + (size_t)(k0 + r) * NEXP + c4);
            *(v4f*)(&Bs[buf][r * B_STRIDE + c4]) = v;   // 16B aligned
        }
        __syncthreads();
        const float* Ab = As[buf];
        const float* Bb = Bs[buf];
#endif
        // ---- compute: 8 k-steps of 4 over the panel ----
        #pragma unroll
        for (int kk = 0; kk < KB; kk += 4) {
            // B fragments: VGPR0 = B[K=kk+2*lhalf][N=lmod], VGPR1 = K+1
            v2f bfrag[2];
            #pragma unroll
            for (int ct = 0; ct < 2; ++ct) {
                const int col = col0 + ct * 16 + lmod;
                bfrag[ct].x = Bb[(kk + 2 * lhalf + 0) * B_STRIDE + col];
                bfrag[ct].y = Bb[(kk + 2 * lhalf + 1) * B_STRIDE + col];
            }
            #pragma unroll
            for (int rt = 0; rt < 4; ++rt) {
                // A fragment: VGPR0 = A[M=lmod][K=kk+2*lhalf], VGPR1 = K+1
                const int r = rt * 16 + lmod;
                v2f afrag;
                afrag.x = Ab[r * A_STRIDE + kk + 2 * lhalf + 0];
                afrag.y = Ab[r * A_STRIDE + kk + 2 * lhalf + 1];
                acc[rt][0] = __builtin_amdgcn_wmma_f32_16x16x4_f32(
                    false, afrag, false, bfrag[0], (short)0, acc[rt][0], false, false);
                acc[rt][1] = __builtin_amdgcn_wmma_f32_16x16x4_f32(
                    false, afrag, false, bfrag[1], (short)0, acc[rt][1], false, false);
            }
        }
    }

    // Epilogue: sigmoid + store. C layout: element (v, lane) -> M = v + 8*lhalf, N = lmod
    #pragma unroll
    for (int rt = 0; rt < 4; ++rt) {
        #pragma unroll
        for (int ct = 0; ct < 2; ++ct) {
            #pragma unroll
            for (int v = 0; v < 8; ++v) {
                const int row = row0 + rt * 16 + v + 8 * lhalf;
                const int col = col0 + ct * 16 + lmod;
                const float logit = acc[rt][ct][v];
                scores[(size_t)row * NEXP + col] = 1.0f / (1.0f + __expf(-logit));
            }
        }
    }
}

// ---------------------------------------------------------------------------
// Kernel 2: routing. One wave32 per row; lane owns experts [lane*8, lane*8+8)
// Group g = experts [32g, 32g+32) spans lanes 4g..4g+3.
// ---------------------------------------------------------------------------
__global__ __launch_bounds__(256) void moe_route_kernel(
    const float* __restrict__ scores, const float* __restrict__ bias,
    int* __restrict__ out_idx, float* __restrict__ out_w)
{
    const int row  = blockIdx.x * 8 + (threadIdx.x >> 5);
    const int lane = threadIdx.x & 31;
    if (row >= NROWS) return;

    const float NEG = -3.402823466e38f;
    const float* srow = scores + (size_t)row * NEXP;

    float sv[8];   // raw sigmoid scores (weights come from these)
    float sr[8];   // scores + bias (routing)
    #pragma unroll
    for (int j = 0; j < 8; ++j) {
        const float s = srow[lane * 8 + j];
        sv[j] = s;
        sr[j] = s + bias[lane * 8 + j];
    }

    // lane-local top-2
    float m1 = NEG, m2 = NEG;
    #pragma unroll
    for (int j = 0; j < 8; ++j) {
        const float v = sr[j];
        if (v > m1) { m2 = m1; m1 = v; }
        else if (v > m2) { m2 = v; }
    }
    // merge top-2 across the 4 lanes of the group
    #pragma unroll
    for (int off = 1; off <= 2; off <<= 1) {
        const float o1 = __shfl_xor(m1, off, 32);
        const float o2 = __shfl_xor(m2, off, 32);
        const float nm1 = fmaxf(m1, o1);
        const float nm2 = fmaxf(fminf(m1, o1), fmaxf(m2, o2));
        m1 = nm1; m2 = nm2;
    }
    const float gscore = m1 + m2;  // group score, valid on all 4 lanes of the group

    // gather all 8 group scores
    float gs[8];
    #pragma unroll
    for (int g = 0; g < 8; ++g) gs[g] = __shfl(gscore, 4 * g, 32);

    // top-4 groups, lowest-index tie-break (matches jax top_k)
    bool gsel[8];
    #pragma unroll
    for (int g = 0; g < 8; ++g) gsel[g] = false;
    for (int t = 0; t < TOPKG; ++t) {
        float best = NEG; int bg = 0;
        #pragma unroll
        for (int g = 0; g < 8; ++g)
            if (!gsel[g] && gs[g] > best) { best = gs[g]; bg = g; }
        gsel[bg] = true;
    }
    const bool active = gsel[lane >> 2];

    float mr[8];
    #pragma unroll
    for (int j = 0; j < 8; ++j) mr[j] = active ? sr[j] : NEG;

    // top-8 experts via 8 wave-wide argmax passes
    unsigned used = 0u;
    int   my_i = 0;
    float my_w = 0.f;
    float wsum = 0.f;
    for (int t = 0; t < TOPK; ++t) {
        float bv = NEG; int lj = -1;
        #pragma unroll
        for (int j = 0; j < 8; ++j)
            if (!((used >> j) & 1u) && mr[j] > bv) { bv = mr[j]; lj = j; }
        int bi = (lj < 0) ? 0x7FFFFFFF : (lane * 8 + lj);
        if (lj < 0) bv = NEG;
        #pragma unroll
        for (int off = 16; off >= 1; off >>= 1) {
            const float ov = __shfl_xor(bv, off, 32);
            const int   oi = __shfl_xor(bi, off, 32);
            if (ov > bv || (ov == bv && oi < bi)) { bv = ov; bi = oi; }
        }
        // bi is uniform across the wave; fetch raw score from owner lane
        const float w = __shfl(sv[bi & 7], bi >> 3, 32);
        wsum += w;
        if (lane == t) { my_i = bi; my_w = w; }
        if ((bi >> 3) == lane) used |= 1u << (bi & 7);
    }

    const float scale = 2.5f / (wsum + 1e-20f);
    if (lane < TOPK) {
        out_idx[(size_t)row * TOPK + lane] = my_i;
        out_w  [(size_t)row * TOPK + lane] = my_w * scale;
    }
}

// ---------------------------------------------------------------------------
extern "C" void kernel_launch(void* const* d_in, const int* in_sizes, int n_in,
                              void* d_out, int out_size, void* d_ws, size_t ws_size,
                              hipStream_t stream) {
    const float* x    = (const float*)d_in[0];   // [8192, 7168]
    const float* W    = (const float*)d_in[1];   // [7168, 256]
    const float* bias = (const float*)d_in[2];   // [256]

    // d_out = [topk_indices int32 8192*8 | topk_weights f32 8192*8 | scores f32 8192*256]
    int*   out_idx    = (int*)d_out;
    float* out_w      = (float*)d_out + (size_t)NROWS * TOPK;
    float* out_scores = (float*)d_out + (size_t)NROWS * TOPK * 2;

    dim3 gemm_grid(NROWS / TILE_ROWS);   // 128 blocks
    dim3 gemm_block(256);
    moe_gemm_sigmoid_kernel<<<gemm_grid, gemm_block, 0, stream>>>(x, W, out_scores);

    dim3 route_grid(NROWS / 8);          // 1024 blocks, 8 rows (waves) each
    dim3 route_block(256);
    moe_route_kernel<<<route_grid, route_block, 0, stream>>>(out_scores, bias, out_idx, out_w);
}